// SGConv_26216480375299
// MI455X (gfx1250) — compile-verified
//
#include <hip/hip_runtime.h>

typedef float v2f __attribute__((ext_vector_type(2)));
typedef float v8f __attribute__((ext_vector_type(8)));

#define IN_F  64
#define OUT_F 128

// ---------------- degree: one thread per edge, u32 atomic inc ----------------
__global__ __launch_bounds__(256) void k_degree(const int* __restrict__ dst,
                                                unsigned* __restrict__ deg,
                                                int n_edges) {
    int e = blockIdx.x * blockDim.x + threadIdx.x;
    if (e < n_edges) atomicAdd(&deg[dst[e]], 1u);
}

// ---------------- norm = deg>0 ? rsqrt(deg) : 0 ----------------
__global__ __launch_bounds__(256) void k_norm(const unsigned* __restrict__ deg,
                                              float* __restrict__ norm, int n) {
    int i = blockIdx.x * blockDim.x + threadIdx.x;
    if (i < n) {
        unsigned d = deg[i];
        norm[i] = d ? rsqrtf((float)d) : 0.0f;
    }
}

// ---------------- row scale: out = in * norm^pw (pw = 1 or 2) ----------------
// one thread per float2 pair; 32 pairs per node row
__global__ __launch_bounds__(256) void k_scale(const float* __restrict__ in,
                                               const float* __restrict__ norm,
                                               float* __restrict__ out,
                                               int n_nodes, int pw) {
    int idx  = blockIdx.x * blockDim.x + threadIdx.x;
    int node = idx >> 5;
    if (node >= n_nodes) return;
    float nn = norm[node];
    if (pw == 2) nn *= nn;
    int p = idx & 31;
    const v2f* ip = (const v2f*)(in  + (size_t)node * IN_F);
    v2f*       op = (v2f*)      (out + (size_t)node * IN_F);
    v2f v = ip[p];
    v.x *= nn; v.y *= nn;
    op[p] = v;
}

// ---------------- SpMM scatter: one wave per edge, 2 feats per lane ----------
// src/dst broadcast to SGPRs via readfirstlane -> coalesced saddr-based vmem.
// Non-returning atomicAdd -> global_atomic_add_f32 (L2 atomic units).
__global__ __launch_bounds__(256) void k_scatter(const float* __restrict__ in,
                                                 float* __restrict__ out,
                                                 const int* __restrict__ src,
                                                 const int* __restrict__ dst,
                                                 int n_edges) {
    int wave = blockIdx.x * (blockDim.x >> 5) + (threadIdx.x >> 5);
    if (wave >= n_edges) return;
    int lane = threadIdx.x & 31;
    int s = __builtin_amdgcn_readfirstlane(src[wave]);
    int d = __builtin_amdgcn_readfirstlane(dst[wave]);
    v2f v = *(const v2f*)(in + (size_t)s * IN_F + lane * 2);
    float* o = out + (size_t)d * IN_F + lane * 2;
    atomicAdd(o,     v.x);
    atomicAdd(o + 1, v.y);
}

// ---------------- GEMM: out[65536,128] = A[65536,64] @ W[128,64]^T + b -------
// One wave per 16x16 output tile via V_WMMA_F32_16X16X4_F32 (16 steps of K=4).
// A-matrix (16x4 f32): lanes 0-15 -> K={0,1}, lanes 16-31 -> K={2,3}, M=lane&15
// B-matrix (4x16 f32): same pattern with N=lane&15 (B[k][n] = W[n][k])
// C/D (16x16 f32, 8 VGPRs): VGPR j -> M = j + 8*(lane>>4), N = lane&15
__global__ __launch_bounds__(256) void k_gemm(const float* __restrict__ A,
                                              const float* __restrict__ W,
                                              const float* __restrict__ bias,
                                              float* __restrict__ out) {
    int wid  = threadIdx.x >> 5;   // 0..7 -> N tile (8*16 = 128 out feats)
    int lane = threadIdx.x & 31;
    int grp  = lane >> 4;          // 0 or 1
    int r    = lane & 15;
    int m_base = blockIdx.x * 16;
    int n_base = wid * 16;

    const float* Arow = A + (size_t)(m_base + r) * IN_F;  // feature row M
    const float* Brow = W + (size_t)(n_base + r) * IN_F;  // weight row N

    v8f c = {0.f, 0.f, 0.f, 0.f, 0.f, 0.f, 0.f, 0.f};
#pragma unroll
    for (int k0 = 0; k0 < IN_F; k0 += 4) {
        v2f a  = *(const v2f*)(Arow + k0 + 2 * grp);
        v2f bb = *(const v2f*)(Brow + k0 + 2 * grp);
        c = __builtin_amdgcn_wmma_f32_16x16x4_f32(
                /*neg_a=*/false, a, /*neg_b=*/false, bb,
                /*c_mod=*/(short)0, c, /*reuse_a=*/false, /*reuse_b=*/false);
    }

    float bv = bias[n_base + r];
    float* ocol = out + (size_t)(m_base + 8 * grp) * OUT_F + n_base + r;
#pragma unroll
    for (int j = 0; j < 8; ++j) {
        ocol[(size_t)j * OUT_F] = c[j] + bv;
    }
}

// ---------------- host-side orchestration ----------------
extern "C" void kernel_launch(void* const* d_in, const int* in_sizes, int n_in,
                              void* d_out, int out_size, void* d_ws, size_t ws_size,
                              hipStream_t stream) {
    const float* feat = (const float*)d_in[0];   // [N, 64]
    const float* W    = (const float*)d_in[1];   // [128, 64]
    const float* bias = (const float*)d_in[2];   // [128]
    const int*   src  = (const int*)d_in[3];     // [E]
    const int*   dst  = (const int*)d_in[4];     // [E]
    float* out = (float*)d_out;                  // [N, 128]

    int n_nodes = in_sizes[0] / IN_F;
    int n_edges = in_sizes[3];

    // workspace layout: norm[N] f32 | deg[N] u32 | bufA[N*64] | bufB[N*64]
    float*    norm = (float*)d_ws;
    unsigned* deg  = (unsigned*)(norm + n_nodes);
    float*    bufA = (float*)(deg + n_nodes);
    float*    bufB = bufA + (size_t)n_nodes * IN_F;

    const size_t featBytes = (size_t)n_nodes * IN_F * sizeof(float);
    const int pair_threads = n_nodes * 32;
    const int edge_waves_grid = (n_edges + 7) / 8;   // 8 waves (edges) per 256-thr block

    // degrees + norm
    hipMemsetAsync(deg, 0, n_nodes * sizeof(unsigned), stream);
    k_degree<<<(n_edges + 255) / 256, 256, 0, stream>>>(dst, deg, n_edges);
    k_norm<<<(n_nodes + 255) / 256, 256, 0, stream>>>(deg, norm, n_nodes);

    // hop 1: pre-norm, scatter
    k_scale<<<(pair_threads + 255) / 256, 256, 0, stream>>>(feat, norm, bufA, n_nodes, 1);
    hipMemsetAsync(bufB, 0, featBytes, stream);
    k_scatter<<<edge_waves_grid, 256, 0, stream>>>(bufA, bufB, src, dst, n_edges);

    // fused post-norm(hop1) * pre-norm(hop2) = norm^2, scatter hop 2
    k_scale<<<(pair_threads + 255) / 256, 256, 0, stream>>>(bufB, norm, bufA, n_nodes, 2);
    hipMemsetAsync(bufB, 0, featBytes, stream);
    k_scatter<<<edge_waves_grid, 256, 0, stream>>>(bufA, bufB, src, dst, n_edges);

    // post-norm hop 2
    k_scale<<<(pair_threads + 255) / 256, 256, 0, stream>>>(bufB, norm, bufA, n_nodes, 1);

    // dense projection with fp32 WMMA
    k_gemm<<<n_nodes / 16, 256, 0, stream>>>(bufA, W, bias, out);
}